// EPnP_4380866642411
// MI455X (gfx1250) — compile-verified
//
#include <hip/hip_runtime.h>
#include <math.h>

// ---------------------------------------------------------------------------
// EPnP for MI455X (gfx1250): one wave32 per batch problem (B=4096, N=256).
// The 512x12 -> 12x12 Gram (M^T M) is computed with V_WMMA_F32_16X16X4_F32,
// with branch-free operand construction so EXEC stays all-1s in the MMA loop.
// ---------------------------------------------------------------------------

typedef float v2f __attribute__((ext_vector_type(2)));
typedef float v8f __attribute__((ext_vector_type(8)));

#define DEVINL static __device__ __forceinline__

DEVINL float wred(float v) {
  v += __shfl_xor(v, 16, 32);
  v += __shfl_xor(v, 8, 32);
  v += __shfl_xor(v, 4, 32);
  v += __shfl_xor(v, 2, 32);
  v += __shfl_xor(v, 1, 32);
  return v;
}

DEVINL float sgnf(float x) { return (x > 0.f) ? 1.f : ((x < 0.f) ? -1.f : 0.f); }
DEVINL float dot3p(const float* a, const float* b) { return a[0]*b[0] + a[1]*b[1] + a[2]*b[2]; }

// symmetric 3x3 eigendecomposition (cyclic Jacobi); V columns = eigenvectors
DEVINL void eig3(float a00, float a01, float a02, float a11, float a12, float a22,
                 float V[3][3], float d[3]) {
  float A[3][3] = {{a00, a01, a02}, {a01, a11, a12}, {a02, a12, a22}};
  for (int i = 0; i < 3; i++)
    for (int j = 0; j < 3; j++) V[i][j] = (i == j) ? 1.f : 0.f;
  for (int sweep = 0; sweep < 10; sweep++) {
    for (int p = 0; p < 2; p++) {
      for (int q = p + 1; q < 3; q++) {
        float apq = A[p][q];
        float aa = fabsf(apq);
        float tau = (A[q][q] - A[p][p]) * 0.5f / ((aa > 1e-30f) ? apq : 1.f);
        float tt = copysignf(1.f, tau) / (fabsf(tau) + sqrtf(1.f + tau * tau));
        float c = 1.f / sqrtf(1.f + tt * tt);
        float s = tt * c;
        if (aa <= 1e-30f) { c = 1.f; s = 0.f; }
        for (int j = 0; j < 3; j++) {
          float gp = A[j][p], gq = A[j][q];
          A[j][p] = c * gp - s * gq; A[j][q] = s * gp + c * gq;
        }
        for (int j = 0; j < 3; j++) {
          float hp = A[p][j], hq = A[q][j];
          A[p][j] = c * hp - s * hq; A[q][j] = s * hp + c * hq;
        }
        for (int j = 0; j < 3; j++) {
          float vp = V[j][p], vq = V[j][q];
          V[j][p] = c * vp - s * vq; V[j][q] = s * vp + c * vq;
        }
      }
    }
  }
  d[0] = A[0][0]; d[1] = A[1][1]; d[2] = A[2][2];
}

DEVINL void sortEig3Desc(float V[3][3], float d[3]) {
  for (int i = 0; i < 2; i++)
    for (int j = i + 1; j < 3; j++)
      if (d[j] > d[i]) {
        float td = d[i]; d[i] = d[j]; d[j] = td;
        for (int r = 0; r < 3; r++) { float tv = V[r][i]; V[r][i] = V[r][j]; V[r][j] = tv; }
      }
}

DEVINL float det3(const float M[3][3]) {
  return M[0][0] * (M[1][1] * M[2][2] - M[1][2] * M[2][1])
       - M[0][1] * (M[1][0] * M[2][2] - M[1][2] * M[2][0])
       + M[0][2] * (M[1][0] * M[2][1] - M[1][1] * M[2][0]);
}

DEVINL void inv4(const float M[4][4], float Vout[4][4]) {
  float a[4][8];
  for (int i = 0; i < 4; i++)
    for (int j = 0; j < 4; j++) { a[i][j] = M[i][j]; a[i][j + 4] = (i == j) ? 1.f : 0.f; }
  for (int k = 0; k < 4; k++) {
    int piv = k; float mx = fabsf(a[k][k]);
    for (int r = k + 1; r < 4; r++) { float vv = fabsf(a[r][k]); if (vv > mx) { mx = vv; piv = r; } }
    if (piv != k)
      for (int j = 0; j < 8; j++) { float ts = a[k][j]; a[k][j] = a[piv][j]; a[piv][j] = ts; }
    float dd = a[k][k]; dd = (fabsf(dd) > 1e-30f) ? dd : 1e-30f;
    float inv = 1.f / dd;
    for (int j = 0; j < 8; j++) a[k][j] *= inv;
    for (int r = 0; r < 4; r++) {
      if (r == k) continue;
      float f = a[r][k];
      for (int j = 0; j < 8; j++) a[r][j] -= f * a[k][j];
    }
  }
  for (int i = 0; i < 4; i++)
    for (int j = 0; j < 4; j++) Vout[i][j] = a[i][j + 4];
}

template <int N>
__device__ __forceinline__ void gsolve(float A[N][N], float b[N], float x[N]) {
  for (int k = 0; k < N; k++) {
    int piv = k; float mx = fabsf(A[k][k]);
    for (int r = k + 1; r < N; r++) { float v = fabsf(A[r][k]); if (v > mx) { mx = v; piv = r; } }
    if (piv != k) {
      for (int c = k; c < N; c++) { float t = A[k][c]; A[k][c] = A[piv][c]; A[piv][c] = t; }
      float t = b[k]; b[k] = b[piv]; b[piv] = t;
    }
    float d = A[k][k]; d = (fabsf(d) > 1e-20f) ? d : 1e-20f;
    float inv = 1.f / d;
    for (int r = k + 1; r < N; r++) {
      float f = A[r][k] * inv;
      for (int c = k; c < N; c++) A[r][c] -= f * A[k][c];
      b[r] -= f * b[k];
    }
  }
  for (int k = N - 1; k >= 0; k--) {
    float s = b[k];
    for (int c = k + 1; c < N; c++) s -= A[k][c] * x[c];
    float d = A[k][k]; d = (fabsf(d) > 1e-20f) ? d : 1e-20f;
    x[k] = s / d;
  }
}

// R = V * diag(1,1,d) * U^T   for  H = U S V^T  (src->tgt rigid fit)
DEVINL void procrustes_R(const float H[3][3], float R[3][3]) {
  float b00 = 0, b01 = 0, b02 = 0, b11 = 0, b12 = 0, b22 = 0;
  for (int k = 0; k < 3; k++) {
    b00 += H[k][0] * H[k][0]; b01 += H[k][0] * H[k][1]; b02 += H[k][0] * H[k][2];
    b11 += H[k][1] * H[k][1]; b12 += H[k][1] * H[k][2]; b22 += H[k][2] * H[k][2];
  }
  float W[3][3], e[3];
  eig3(b00, b01, b02, b11, b12, b22, W, e);
  sortEig3Desc(W, e);
  float u0c[3], u1c[3];
  for (int r = 0; r < 3; r++) {
    u0c[r] = H[r][0] * W[0][0] + H[r][1] * W[1][0] + H[r][2] * W[2][0];
    u1c[r] = H[r][0] * W[0][1] + H[r][1] * W[1][1] + H[r][2] * W[2][1];
  }
  float n0 = sqrtf(dot3p(u0c, u0c));
  if (n0 < 1e-20f) { u0c[0] = 1.f; u0c[1] = 0.f; u0c[2] = 0.f; n0 = 1.f; }
  float in0 = 1.f / n0;
  u0c[0] *= in0; u0c[1] *= in0; u0c[2] *= in0;
  float dp = dot3p(u1c, u0c);
  u1c[0] -= dp * u0c[0]; u1c[1] -= dp * u0c[1]; u1c[2] -= dp * u0c[2];
  float n1 = sqrtf(dot3p(u1c, u1c));
  if (n1 < 1e-12f) {
    float ax = fabsf(u0c[0]);
    float e0 = (ax < 0.9f) ? 1.f : 0.f;
    float e1 = 1.f - e0;
    u1c[0] = -u0c[2] * e1;
    u1c[1] = u0c[2] * e0;
    u1c[2] = u0c[0] * e1 - u0c[1] * e0;
    n1 = sqrtf(dot3p(u1c, u1c));
    if (n1 < 1e-20f) n1 = 1.f;
  }
  float in1 = 1.f / n1;
  u1c[0] *= in1; u1c[1] *= in1; u1c[2] *= in1;
  float u2c[3] = { u0c[1] * u1c[2] - u0c[2] * u1c[1],
                   u0c[2] * u1c[0] - u0c[0] * u1c[2],
                   u0c[0] * u1c[1] - u0c[1] * u1c[0] };
  float U[3][3];
  for (int r = 0; r < 3; r++) { U[r][0] = u0c[r]; U[r][1] = u1c[r]; U[r][2] = u2c[r]; }
  float dfix = (det3(W) >= 0.f) ? 1.f : -1.f;   // det(U)=+1 by construction
  for (int a = 0; a < 3; a++)
    for (int c = 0; c < 3; c++)
      R[a][c] = W[a][0] * U[c][0] + W[a][1] * U[c][1] + dfix * W[a][2] * U[c][2];
}

// one EPnP candidate: camera control points, scale, sign, Procrustes, reproj err
static __device__ void eval_solution(
    const float beta[4], const float* sNull, const float* sAlpha,
    const float px[8][3], const float pu[8], const float pv[8],
    const float cw[3], float fu, float fv, float u0, float v0,
    int lane, bool wantErr,
    float R[3][3], float t[3], float* scaleOut, float* errOut) {
  float bc[4][3];
  for (int k = 0; k < 4; k++)
    for (int c = 0; c < 3; c++) {
      float s = 0.f;
      for (int m = 0; m < 4; m++) s += beta[m] * sNull[m * 12 + k * 3 + c];
      bc[k][c] = s;
    }
  float tp[8][3];
  float st0 = 0.f, st1 = 0.f, st2 = 0.f;
  for (int ii = 0; ii < 8; ii++) {
    int i = ii * 32 + lane;
    float a0 = sAlpha[i * 4 + 0], a1 = sAlpha[i * 4 + 1];
    float a2 = sAlpha[i * 4 + 2], a3 = sAlpha[i * 4 + 3];
    for (int c = 0; c < 3; c++)
      tp[ii][c] = a0 * bc[0][c] + a1 * bc[1][c] + a2 * bc[2][c] + a3 * bc[3][c];
    st0 += tp[ii][0]; st1 += tp[ii][1]; st2 += tp[ii][2];
  }
  float ct[3];
  ct[0] = wred(st0) * (1.f / 256.f);
  ct[1] = wred(st1) * (1.f / 256.f);
  ct[2] = wred(st2) * (1.f / 256.f);

  float sncw = 0.f, sncc = 0.f;
  for (int ii = 0; ii < 8; ii++) {
    float d0 = tp[ii][0] - ct[0], d1 = tp[ii][1] - ct[1], d2 = tp[ii][2] - ct[2];
    float w0 = px[ii][0] - cw[0], w1 = px[ii][1] - cw[1], w2 = px[ii][2] - cw[2];
    float nc = sqrtf(d0 * d0 + d1 * d1 + d2 * d2);
    float nw = sqrtf(w0 * w0 + w1 * w1 + w2 * w2);
    sncw += nc * nw; sncc += nc * nc;
  }
  sncw = wred(sncw); sncc = wred(sncc);
  float scale = sncw / fmaxf(sncc, 1e-30f);
  float sg = (scale * ct[2] < 0.f) ? -1.f : 1.f;
  float s2 = scale * sg;
  float ct2[3] = { ct[0] * s2, ct[1] * s2, ct[2] * s2 };

  float Hm[3][3] = {{0, 0, 0}, {0, 0, 0}, {0, 0, 0}};
  for (int ii = 0; ii < 8; ii++) {
    float w0 = px[ii][0] - cw[0], w1 = px[ii][1] - cw[1], w2 = px[ii][2] - cw[2];
    float c0 = tp[ii][0] * s2 - ct2[0], c1 = tp[ii][1] * s2 - ct2[1], c2 = tp[ii][2] * s2 - ct2[2];
    Hm[0][0] += w0 * c0; Hm[0][1] += w0 * c1; Hm[0][2] += w0 * c2;
    Hm[1][0] += w1 * c0; Hm[1][1] += w1 * c1; Hm[1][2] += w1 * c2;
    Hm[2][0] += w2 * c0; Hm[2][1] += w2 * c1; Hm[2][2] += w2 * c2;
  }
  for (int a = 0; a < 3; a++)
    for (int c = 0; c < 3; c++) Hm[a][c] = wred(Hm[a][c]);

  procrustes_R(Hm, R);
  for (int a = 0; a < 3; a++)
    t[a] = ct2[a] - (R[a][0] * cw[0] + R[a][1] * cw[1] + R[a][2] * cw[2]);
  *scaleOut = s2;

  if (wantErr) {
    float es = 0.f;
    for (int ii = 0; ii < 8; ii++) {
      float pcx = R[0][0] * px[ii][0] + R[0][1] * px[ii][1] + R[0][2] * px[ii][2] + t[0];
      float pcy = R[1][0] * px[ii][0] + R[1][1] * px[ii][1] + R[1][2] * px[ii][2] + t[1];
      float pcz = R[2][0] * px[ii][0] + R[2][1] * px[ii][1] + R[2][2] * px[ii][2] + t[2];
      float iz = 1.f / pcz;
      float uu = (fu * pcx + u0 * pcz) * iz;
      float vv = (fv * pcy + v0 * pcz) * iz;
      float eu = uu - pu[ii], evv = vv - pv[ii];
      es += sqrtf(eu * eu + evv * evv);
    }
    *errOut = wred(es) * (1.f / 256.f);
  }
}

__global__ __launch_bounds__(32) void epnp_kernel(
    const float* __restrict__ points, const float* __restrict__ pixels,
    const float* __restrict__ Kin, float* __restrict__ out) {
  const int b = blockIdx.x;
  const int lane = threadIdx.x;

  __shared__ float sAlpha[1024];  // (256,4)
  __shared__ float sDu[256];      // u0-u
  __shared__ float sDv[256];      // v0-v
  __shared__ float sG[256];       // 16x16 WMMA Gram tile
  __shared__ float sA[144];       // 12x12 Jacobi work matrix
  __shared__ float sV[144];       // 12x12 eigenvectors
  __shared__ float sNull[48];     // 4 x 12 null-space vectors

  const float* P = points + (size_t)b * 768;
  const float* X = pixels + (size_t)b * 512;
  const float* Kb = Kin + (size_t)b * 9;
  const float fu = Kb[0], u0 = Kb[2], fv = Kb[4], v0 = Kb[5];

  // ---- load points/pixels (8 per lane, coalesced), centroid + covariance ----
  float px[8][3], pu[8], pv[8];
  float sx = 0.f, sy = 0.f, sz = 0.f;
  for (int ii = 0; ii < 8; ii++) {
    int i = ii * 32 + lane;
    px[ii][0] = P[i * 3 + 0]; px[ii][1] = P[i * 3 + 1]; px[ii][2] = P[i * 3 + 2];
    pu[ii] = X[i * 2 + 0]; pv[ii] = X[i * 2 + 1];
    sDu[i] = u0 - pu[ii]; sDv[i] = v0 - pv[ii];
    sx += px[ii][0]; sy += px[ii][1]; sz += px[ii][2];
  }
  float cw[3];
  cw[0] = wred(sx) * (1.f / 256.f);
  cw[1] = wred(sy) * (1.f / 256.f);
  cw[2] = wred(sz) * (1.f / 256.f);

  float cxx = 0, cxy = 0, cxz = 0, cyy = 0, cyz = 0, czz = 0;
  for (int ii = 0; ii < 8; ii++) {
    float dx = px[ii][0] - cw[0], dy = px[ii][1] - cw[1], dz = px[ii][2] - cw[2];
    cxx += dx * dx; cxy += dx * dy; cxz += dx * dz;
    cyy += dy * dy; cyz += dy * dz; czz += dz * dz;
  }
  cxx = wred(cxx); cxy = wred(cxy); cxz = wred(cxz);
  cyy = wred(cyy); cyz = wred(cyz); czz = wred(czz);

  // ---- SVD basis (control points) ----
  float W3[3][3], ev[3];
  eig3(cxx, cxy, cxz, cyy, cyz, czz, W3, ev);
  sortEig3Desc(W3, ev);
  float bb[4][3];
  bb[0][0] = cw[0]; bb[0][1] = cw[1]; bb[0][2] = cw[2];
  for (int i = 0; i < 3; i++) {
    float sc = sqrtf(fmaxf(ev[i], 0.f));
    bb[i + 1][0] = cw[0] + sc * W3[i][0];
    bb[i + 1][1] = cw[1] + sc * W3[i][1];
    bb[i + 1][2] = cw[2] + sc * W3[i][2];
  }

  // ---- alpha = points_h @ inv(bases_h) ----
  float Bh[4][4], Binv[4][4];
  for (int r = 0; r < 4; r++) { Bh[r][0] = bb[r][0]; Bh[r][1] = bb[r][1]; Bh[r][2] = bb[r][2]; Bh[r][3] = 1.f; }
  inv4(Bh, Binv);
  for (int ii = 0; ii < 8; ii++) {
    int i = ii * 32 + lane;
    for (int k = 0; k < 4; k++)
      sAlpha[i * 4 + k] = px[ii][0] * Binv[0][k] + px[ii][1] * Binv[1][k] +
                          px[ii][2] * Binv[2][k] + Binv[3][k];
  }
  __syncthreads();

  // ---- Gram of M (512x12) via V_WMMA_F32_16X16X4_F32 -------------------
  // For a Gram matrix the A (16x4) and B (4x16) per-lane register images are
  // identical: lane (lh*16+c) holds M[k0+2*lh+j][c] in element j.
  // Branch-free operand build: selection conditions are loop-invariant per
  // lane, folded into three coefficients so EXEC stays all-1s.
  const int laneHalf = lane >> 4;
  const int cidx = lane & 15;
  const bool cvalid = (cidx < 12);
  const int kk = cvalid ? (cidx / 3) : 0;
  const int ssel = cvalid ? (cidx - kk * 3) : 0;
  const float c0a = (cvalid && ssel == 0) ? fu : 0.f;  // u-row: alpha*fu
  const float c1a = (cvalid && ssel == 1) ? fv : 0.f;  // v-row: alpha*fv
  const float cd  = (cvalid && ssel == 2) ? 1.f : 0.f; // alpha*(u0-u)/(v0-v)

  v8f acc0 = {}, acc1 = {}, acc2 = {}, acc3 = {};
  for (int t4 = 0; t4 < 32; t4++) {
    int chBase = t4 * 4;
#define EPNP_WMMA_STEP(ACC, CH)                                                 \
    {                                                                           \
      int ch = (CH);                                                            \
      int ip = ch * 2 + laneHalf;                                               \
      float aV = sAlpha[ip * 4 + kk];                                           \
      float du = sDu[ip];                                                       \
      float dv = sDv[ip];                                                       \
      float m0 = fmaf(aV * du, cd, aV * c0a);                                   \
      float m1 = fmaf(aV * dv, cd, aV * c1a);                                   \
      v2f ab; ab[0] = m0; ab[1] = m1;                                           \
      ACC = __builtin_amdgcn_wmma_f32_16x16x4_f32(false, ab, false, ab,         \
                                                  (short)0, ACC, false, false); \
    }
    EPNP_WMMA_STEP(acc0, chBase + 0)
    EPNP_WMMA_STEP(acc1, chBase + 1)
    EPNP_WMMA_STEP(acc2, chBase + 2)
    EPNP_WMMA_STEP(acc3, chBase + 3)
#undef EPNP_WMMA_STEP
  }
  v8f accT = acc0 + acc1 + acc2 + acc3;
  {
    int row0 = (lane < 16) ? 0 : 8;
    for (int v = 0; v < 8; v++) sG[(row0 + v) * 16 + cidx] = accT[v];
  }
  __syncthreads();

  // ---- 12x12 symmetric eigensolver (cooperative cyclic Jacobi) ----
  for (int e = lane; e < 144; e += 32) {
    int r = e / 12, c = e % 12;
    sA[e] = sG[r * 16 + c];
    sV[e] = (r == c) ? 1.f : 0.f;
  }
  __syncthreads();
  for (int sweep = 0; sweep < 8; sweep++) {
    for (int p = 0; p < 11; p++) {
      for (int q = p + 1; q < 12; q++) {
        float app = sA[p * 12 + p], aqq = sA[q * 12 + q], apq = sA[p * 12 + q];
        float aa = fabsf(apq);
        float tau = (aqq - app) * 0.5f / ((aa > 1e-30f) ? apq : 1.f);
        float tt = copysignf(1.f, tau) / (fabsf(tau) + sqrtf(1.f + tau * tau));
        float cth = 1.f / sqrtf(1.f + tt * tt);
        float sth = tt * cth;
        if (aa <= 1e-30f) { cth = 1.f; sth = 0.f; }
        float gp = 0.f, gq = 0.f;
        if (lane < 12) { gp = sA[lane * 12 + p]; gq = sA[lane * 12 + q]; }
        __syncthreads();
        if (lane < 12) { sA[lane * 12 + p] = cth * gp - sth * gq; sA[lane * 12 + q] = sth * gp + cth * gq; }
        __syncthreads();
        float hp = 0.f, hq = 0.f;
        if (lane < 12) { hp = sA[p * 12 + lane]; hq = sA[q * 12 + lane]; }
        __syncthreads();
        if (lane < 12) {
          sA[p * 12 + lane] = cth * hp - sth * hq;
          sA[q * 12 + lane] = sth * hp + cth * hq;
          float vp = sV[lane * 12 + p], vq = sV[lane * 12 + q];
          sV[lane * 12 + p] = cth * vp - sth * vq;
          sV[lane * 12 + q] = sth * vp + cth * vq;
        }
        __syncthreads();
      }
    }
  }

  // ---- pick 4 smallest eigenpairs; nullv rows = [4th,3rd,2nd,1st smallest] ----
  float dloc[12];
  for (int j = 0; j < 12; j++) dloc[j] = sA[j * 12 + j];
  int idx[4];
  for (int tsel = 0; tsel < 4; tsel++) {
    int bi = 0; float bv = dloc[0];
    for (int j = 1; j < 12; j++) if (dloc[j] < bv) { bv = dloc[j]; bi = j; }
    idx[tsel] = bi; dloc[bi] = 3.4e38f;
  }
  __syncthreads();
  for (int e = lane; e < 48; e += 32) {
    int m = e / 12, j = e % 12;
    sNull[m * 12 + j] = sV[j * 12 + idx[3 - m]];
  }
  __syncthreads();

  // ---- l_mat (6x10) and rho (6), redundant per lane ----
  const int LIc[6] = {1, 2, 3, 2, 3, 3};
  const int LJc[6] = {0, 0, 0, 1, 1, 2};
  float dvl[4][6][3];
  for (int m = 0; m < 4; m++)
    for (int p = 0; p < 6; p++)
      for (int c = 0; c < 3; c++)
        dvl[m][p][c] = sNull[m * 12 + LIc[p] * 3 + c] - sNull[m * 12 + LJc[p] * 3 + c];
  const int A10c[10] = {0, 0, 1, 0, 1, 2, 0, 1, 2, 3};
  const int B10c[10] = {0, 1, 1, 2, 2, 2, 3, 3, 3, 3};
  const float multc[10] = {1, 2, 1, 2, 2, 1, 2, 2, 2, 1};
  float Lm[6][10];
  for (int p = 0; p < 6; p++)
    for (int t10 = 0; t10 < 10; t10++)
      Lm[p][t10] = multc[t10] * dot3p(dvl[A10c[t10]][p], dvl[B10c[t10]][p]);
  float rho[6];
  for (int p = 0; p < 6; p++) {
    float d0 = bb[LIc[p]][0] - bb[LJc[p]][0];
    float d1 = bb[LIc[p]][1] - bb[LJc[p]][1];
    float d2 = bb[LIc[p]][2] - bb[LJc[p]][2];
    rho[p] = d0 * d0 + d1 * d1 + d2 * d2;
  }

  // ---- four beta candidates ----
  float betasC[4][4];
  betasC[0][0] = 0.f; betasC[0][1] = 0.f; betasC[0][2] = 0.f; betasC[0][3] = 1.f;
  {  // cols {5,8,9}: 3-var normal equations
    float N3[3][3] = {{0, 0, 0}, {0, 0, 0}, {0, 0, 0}}, r3[3] = {0, 0, 0}, S[3];
    for (int p = 0; p < 6; p++) {
      float l0 = Lm[p][5], l1 = Lm[p][8], l2 = Lm[p][9];
      N3[0][0] += l0 * l0; N3[0][1] += l0 * l1; N3[0][2] += l0 * l2;
      N3[1][1] += l1 * l1; N3[1][2] += l1 * l2; N3[2][2] += l2 * l2;
      r3[0] += l0 * rho[p]; r3[1] += l1 * rho[p]; r3[2] += l2 * rho[p];
    }
    N3[1][0] = N3[0][1]; N3[2][0] = N3[0][2]; N3[2][1] = N3[1][2];
    gsolve<3>(N3, r3, S);
    betasC[1][0] = 0.f; betasC[1][1] = 0.f;
    betasC[1][2] = sqrtf(fabsf(S[0]));
    betasC[1][3] = sqrtf(fabsf(S[2])) * sgnf(S[1]) * sgnf(S[0]);
  }
  {  // cols {2,4,5,7,8,9}: 6x6 direct solve
    float A6[6][6], r6[6], S[6];
    const int cs[6] = {2, 4, 5, 7, 8, 9};
    for (int p = 0; p < 6; p++) {
      for (int j = 0; j < 6; j++) A6[p][j] = Lm[p][cs[j]];
      r6[p] = rho[p];
    }
    gsolve<6>(A6, r6, S);
    betasC[2][0] = 0.f;
    betasC[2][1] = sqrtf(fabsf(S[0]));
    betasC[2][2] = sqrtf(fabsf(S[2])) * sgnf(S[1]) * sgnf(S[0]);
    betasC[2][3] = sqrtf(fabsf(S[5])) * sgnf(S[3]) * sgnf(S[0]);
  }
  {  // full 6x10: min-norm via L L^T
    float G6[6][6], r6[6], y[6], S[10];
    for (int p = 0; p < 6; p++) {
      for (int q = 0; q < 6; q++) {
        float s = 0.f;
        for (int t10 = 0; t10 < 10; t10++) s += Lm[p][t10] * Lm[q][t10];
        G6[p][q] = s;
      }
      r6[p] = rho[p];
    }
    gsolve<6>(G6, r6, y);
    for (int t10 = 0; t10 < 10; t10++) {
      float s = 0.f;
      for (int p = 0; p < 6; p++) s += Lm[p][t10] * y[p];
      S[t10] = s;
    }
    betasC[3][0] = sqrtf(fabsf(S[0]));
    betasC[3][1] = sqrtf(fabsf(S[2])) * sgnf(S[1]) * sgnf(S[0]);
    betasC[3][2] = sqrtf(fabsf(S[5])) * sgnf(S[3]) * sgnf(S[0]);
    betasC[3][3] = sqrtf(fabsf(S[9])) * sgnf(S[6]) * sgnf(S[0]);
  }

  // ---- evaluate candidates, pick argmin reprojection error ----
  float bestErr = 3.4e38f, bestBeta[4] = {0, 0, 0, 1}, bestScale = 1.f;
  float Rtmp[3][3], ttmp[3];
  for (int cnd = 0; cnd < 4; cnd++) {
    float sc, er;
    eval_solution(betasC[cnd], sNull, sAlpha, px, pu, pv, cw, fu, fv, u0, v0,
                  lane, true, Rtmp, ttmp, &sc, &er);
    if (er < bestErr) {
      bestErr = er; bestScale = sc;
      for (int m = 0; m < 4; m++) bestBeta[m] = betasC[cnd][m];
    }
  }
  float bref[4];
  for (int m = 0; m < 4; m++) bref[m] = bestBeta[m] * bestScale;

  // ---- Gauss-Newton refinement (10 iterations), redundant per lane ----
  const int BIc[6] = {0, 0, 0, 1, 1, 2};
  const int BJc[6] = {1, 2, 3, 2, 3, 3};
  float dvb[4][6][3];
  for (int m = 0; m < 4; m++)
    for (int p = 0; p < 6; p++)
      for (int c = 0; c < 3; c++)
        dvb[m][p][c] = sNull[m * 12 + BIc[p] * 3 + c] - sNull[m * 12 + BJc[p] * 3 + c];
  float dwn[6];
  for (int p = 0; p < 6; p++) dwn[p] = sqrtf(rho[p]);
  for (int it = 0; it < 10; it++) {
    float Jac[6][4], rres[6];
    for (int p = 0; p < 6; p++) {
      float d0 = 0.f, d1 = 0.f, d2 = 0.f;
      for (int m = 0; m < 4; m++) {
        d0 += bref[m] * dvb[m][p][0];
        d1 += bref[m] * dvb[m][p][1];
        d2 += bref[m] * dvb[m][p][2];
      }
      float dcn = sqrtf(d0 * d0 + d1 * d1 + d2 * d2);
      rres[p] = dwn[p] - dcn;
      float inv = 1.f / fmaxf(dcn, 1e-12f);
      for (int m = 0; m < 4; m++)
        Jac[p][m] = -(d0 * dvb[m][p][0] + d1 * dvb[m][p][1] + d2 * dvb[m][p][2]) * inv;
    }
    float JtJ[4][4], Jtb[4], xs[4];
    for (int a = 0; a < 4; a++) {
      for (int c = 0; c < 4; c++) {
        float s = 0.f;
        for (int p = 0; p < 6; p++) s += Jac[p][a] * Jac[p][c];
        JtJ[a][c] = s;
      }
      float s = 0.f;
      for (int p = 0; p < 6; p++) s += Jac[p][a] * (-rres[p]);
      Jtb[a] = s;
    }
    gsolve<4>(JtJ, Jtb, xs);
    for (int m = 0; m < 4; m++) bref[m] += xs[m];
  }

  // ---- final solution + quaternion ----
  float Rf[3][3], tf[3], scf, erf;
  eval_solution(bref, sNull, sAlpha, px, pu, pv, cw, fu, fv, u0, v0,
                lane, false, Rf, tf, &scf, &erf);
  float m00 = Rf[0][0], m11 = Rf[1][1], m22 = Rf[2][2];
  float qw = 0.5f * sqrtf(fmaxf(1.f + m00 + m11 + m22, 1e-12f));
  float qx = 0.5f * sqrtf(fmaxf(1.f + m00 - m11 - m22, 1e-12f));
  float qy = 0.5f * sqrtf(fmaxf(1.f - m00 + m11 - m22, 1e-12f));
  float qz = 0.5f * sqrtf(fmaxf(1.f - m00 - m11 + m22, 1e-12f));
  qx *= (Rf[2][1] - Rf[1][2] < 0.f) ? -1.f : 1.f;
  qy *= (Rf[0][2] - Rf[2][0] < 0.f) ? -1.f : 1.f;
  qz *= (Rf[1][0] - Rf[0][1] < 0.f) ? -1.f : 1.f;
  if (lane == 0) {
    float* o = out + (size_t)b * 7;
    o[0] = tf[0]; o[1] = tf[1]; o[2] = tf[2];
    o[3] = qx; o[4] = qy; o[5] = qz; o[6] = qw;
  }
}

extern "C" void kernel_launch(void* const* d_in, const int* in_sizes, int n_in,
                              void* d_out, int out_size, void* d_ws, size_t ws_size,
                              hipStream_t stream) {
  (void)n_in; (void)out_size; (void)d_ws; (void)ws_size;
  const float* points = (const float*)d_in[0];
  const float* pixels = (const float*)d_in[1];
  const float* K = (const float*)d_in[2];
  float* out = (float*)d_out;
  int B = in_sizes[0] / (256 * 3);
  if (B < 1) B = 1;
  epnp_kernel<<<B, 32, 0, stream>>>(points, pixels, K, out);
}